// GCNLayer_49168785605217
// MI455X (gfx1250) — compile-verified
//
#include <hip/hip_runtime.h>

#define N_NODES 100000
#define N_EDGES 625000
#define D_FEAT  128

typedef __attribute__((ext_vector_type(2))) float v2f;
typedef __attribute__((ext_vector_type(4))) float v4f;
typedef __attribute__((ext_vector_type(8))) float v8f;

// ---------------------------------------------------------------------------
// Kernel 1: zero the accumulator h[N_NODES][D_FEAT] with 16B stores.
// 100000*128 floats = 3,200,000 float4 -> 12500 blocks x 256 threads exactly.
// ---------------------------------------------------------------------------
__global__ void gcn_zero_h(float* __restrict__ h) {
    size_t i = (size_t)blockIdx.x * blockDim.x + threadIdx.x;
    v4f z = {};
    ((v4f*)h)[i] = z;
}

// ---------------------------------------------------------------------------
// Kernel 2: edge scatter. One wave32 per edge; lane l owns floats [4l, 4l+4).
// float4 gather from feature[src[e]] (coalesced 16B, L2-resident table),
// then 4 native global_atomic_add_f32 into h[dst[e]] (relaxed, agent scope
// -> hardware fadd, no CAS loop).
// ---------------------------------------------------------------------------
__global__ void gcn_edge_scatter(const float* __restrict__ feat,
                                 const int*   __restrict__ src,
                                 const int*   __restrict__ dst,
                                 float*       __restrict__ h,
                                 int n_edges) {
    int wave = threadIdx.x >> 5;
    int lane = threadIdx.x & 31;
    int e = blockIdx.x * 8 + wave;
    if (e >= n_edges) return;

    int s = src[e];
    int d = dst[e];

    v4f v = ((const v4f*)(feat + (size_t)s * D_FEAT))[lane];
    float* hd = h + (size_t)d * D_FEAT + lane * 4;

    __hip_atomic_fetch_add(hd + 0, v.x, __ATOMIC_RELAXED, __HIP_MEMORY_SCOPE_AGENT);
    __hip_atomic_fetch_add(hd + 1, v.y, __ATOMIC_RELAXED, __HIP_MEMORY_SCOPE_AGENT);
    __hip_atomic_fetch_add(hd + 2, v.z, __ATOMIC_RELAXED, __HIP_MEMORY_SCOPE_AGENT);
    __hip_atomic_fetch_add(hd + 3, v.w, __ATOMIC_RELAXED, __HIP_MEMORY_SCOPE_AGENT);
}

// ---------------------------------------------------------------------------
// Kernel 3: out = h @ W^T + b via V_WMMA_F32_16X16X4_F32 (full fp32 WMMA).
// Block = 256 threads = 8 waves. blockIdx.x -> 16-node strip; wave w -> 16
// output channels [16w, 16w+16). Each wave accumulates a 16x16 f32 tile over
// K = 128 in 32 WMMA steps.
//
// A fragment (16x4 f32, 2 VGPRs): lanes 0-15 hold row M=lane, K={k,k+1};
// lanes 16-31 hold row M=lane-16, K={k+2,k+3}  -> one contiguous float2 load.
// B fragment (4x16 f32, 2 VGPRs): B[k][n] = W[n][k]; same half-wave K split,
// N = lane&15 -> contiguous float2 load from W row-major.
// C/D (8 VGPRs): VGPR r holds M=r (lanes 0-15) / M=r+8 (lanes 16-31), N=lane&15.
// ---------------------------------------------------------------------------
__global__ void gcn_gemm_wmma_bias(const float* __restrict__ h,
                                   const float* __restrict__ W,
                                   const float* __restrict__ bias,
                                   float*       __restrict__ out) {
    int wave = threadIdx.x >> 5;
    int lane = threadIdx.x & 31;
    int half = lane >> 4;    // 0: K pair {0,1}, 1: K pair {2,3}
    int l15  = lane & 15;

    int node0 = blockIdx.x * 16;
    int o0    = wave * 16;

    const float* arow = h + (size_t)(node0 + l15) * D_FEAT + half * 2;
    const float* brow = W + (size_t)(o0    + l15) * D_FEAT + half * 2;

    v8f c = {};
#pragma unroll
    for (int k = 0; k < D_FEAT; k += 4) {
        v2f a = *(const v2f*)(arow + k);
        v2f b = *(const v2f*)(brow + k);
        // (neg_a, A, neg_b, B, c_mod, C, reuse_a, reuse_b)
        c = __builtin_amdgcn_wmma_f32_16x16x4_f32(
                false, a, false, b, (short)0, c, false, false);
    }

    float bb = bias[o0 + l15];
    float* orow = out + (size_t)(node0 + half * 8) * D_FEAT + o0 + l15;
#pragma unroll
    for (int r = 0; r < 8; ++r) {
        orow[(size_t)r * D_FEAT] = c[r] + bb;
    }
}

// ---------------------------------------------------------------------------
// inputs: feature[100000*128] f32, src[625000] int, dst[625000] int,
//         W[128*128] f32, b[128] f32.   out: [100000*128] f32.
// d_ws: h accumulator, 51.2 MB.
// ---------------------------------------------------------------------------
extern "C" void kernel_launch(void* const* d_in, const int* in_sizes, int n_in,
                              void* d_out, int out_size, void* d_ws, size_t ws_size,
                              hipStream_t stream) {
    const float* feat = (const float*)d_in[0];
    const int*   src  = (const int*)  d_in[1];
    const int*   dst  = (const int*)  d_in[2];
    const float* W    = (const float*)d_in[3];
    const float* b    = (const float*)d_in[4];
    float*       out  = (float*)d_out;
    float*       h    = (float*)d_ws;

    // 1) zero accumulator: 3.2M float4 / 256 threads = 12500 blocks
    gcn_zero_h<<<(N_NODES * D_FEAT) / (256 * 4), 256, 0, stream>>>(h);

    // 2) scatter-add: 8 edges (waves) per 256-thread block
    gcn_edge_scatter<<<(N_EDGES + 7) / 8, 256, 0, stream>>>(feat, src, dst, h, N_EDGES);

    // 3) fp32 WMMA GEMM + bias: 100000/16 = 6250 node strips
    gcn_gemm_wmma_bias<<<N_NODES / 16, 256, 0, stream>>>(h, W, b, out);
}